// GNNLayer_23252952940857
// MI455X (gfx1250) — compile-verified
//
#include <hip/hip_runtime.h>

#define DIM 128

typedef float v2f __attribute__((ext_vector_type(2)));
typedef float v8f __attribute__((ext_vector_type(8)));

// ---------------------------------------------------------------------------
// GEMM: P[N x 128] = X[N x 128] @ W[128 x 128], fp32 via V_WMMA_F32_16X16X4_F32
// Block = 256 threads = 8 waves; each wave computes a 16-row x 128-col stripe.
// W is staged in LDS pair-interleaved so each B fragment is one ds_load_b64:
//   W[k][n] lives at Wlds[((k>>1)*DIM + n)*2 + (k&1)]
// ---------------------------------------------------------------------------
__global__ void __launch_bounds__(256)
gemm_xw_wmma(const float* __restrict__ X, const float* __restrict__ W,
             float* __restrict__ P, int N) {
  __shared__ float Wlds[DIM * DIM];  // 64 KB
  const int tid = threadIdx.x;

  // Cooperative stage of W (coalesced float4 loads, scattered b32 LDS stores)
  for (int i = tid; i < DIM * DIM / 4; i += 256) {
    const int k = (i * 4) / DIM;
    const int n = (i * 4) % DIM;
    const float4 w4 = ((const float4*)W)[i];
    const int base = ((k >> 1) * DIM) * 2 + (k & 1);
    Wlds[base + (n + 0) * 2] = w4.x;
    Wlds[base + (n + 1) * 2] = w4.y;
    Wlds[base + (n + 2) * 2] = w4.z;
    Wlds[base + (n + 3) * 2] = w4.w;
  }
  __syncthreads();

  const int lane = tid & 31;
  const int wave = tid >> 5;
  const int half = lane >> 4;   // 0: lanes 0-15, 1: lanes 16-31
  const int ln   = lane & 15;
  const int row0 = (blockIdx.x * 8 + wave) * 16;
  if (row0 >= N) return;        // wave-uniform; EXEC stays all-ones for WMMA

  v8f acc[8];
#pragma unroll
  for (int t = 0; t < 8; ++t) acc[t] = v8f{};

  const float* Xrow = X + (size_t)(row0 + ln) * DIM;

#pragma unroll 2
  for (int kk = 0; kk < DIM; kk += 4) {
    const int ka = kk + half * 2;
    // A fragment: lane holds X[row][ka], X[row][ka+1]  (8B-aligned global load)
    const v2f a = *(const v2f*)(Xrow + ka);
#pragma unroll
    for (int t = 0; t < 8; ++t) {
      const int n = t * 16 + ln;
      // B fragment: W[ka][n], W[ka+1][n] -> contiguous in pair-interleaved LDS
      const v2f b = *(const v2f*)(&Wlds[((ka >> 1) * DIM + n) * 2]);
      acc[t] = __builtin_amdgcn_wmma_f32_16x16x4_f32(
          /*neg_a=*/false, a, /*neg_b=*/false, b,
          /*c_mod=*/(short)0, acc[t], /*reuse_a=*/false, /*reuse_b=*/false);
    }
  }

  // D layout: VGPR j in lane l -> row (j + 8*(l>>4)), col (t*16 + l&15)
  float* Prow = P + (size_t)row0 * DIM;
#pragma unroll
  for (int t = 0; t < 8; ++t)
#pragma unroll
    for (int j = 0; j < 8; ++j)
      Prow[(size_t)(j + half * 8) * DIM + t * 16 + ln] = acc[t][j];
}

// ---------------------------------------------------------------------------
// Degree count: one thread per edge, f32 atomic increments (L2-resident)
// ---------------------------------------------------------------------------
__global__ void __launch_bounds__(256)
count_deg(const int* __restrict__ dst, float* __restrict__ deg, int E) {
  const int stride = gridDim.x * blockDim.x;
  for (int e = blockIdx.x * blockDim.x + threadIdx.x; e < E; e += stride)
    unsafeAtomicAdd(deg + dst[e], 1.0f);
}

// ---------------------------------------------------------------------------
// Edge scatter: one wave per edge; lane handles 4 columns (float4 gather +
// 4x global_atomic_add_f32). Source rows are L2-resident (51.2 MB < 192 MB).
// ---------------------------------------------------------------------------
__global__ void __launch_bounds__(256)
scatter_rows(const float* __restrict__ P, const int* __restrict__ src,
             const int* __restrict__ dst, float* __restrict__ agg, int E) {
  const int lane   = threadIdx.x & 31;
  const int gwave  = (blockIdx.x * blockDim.x + threadIdx.x) >> 5;
  const int nwaves = (gridDim.x * blockDim.x) >> 5;
  for (int e = gwave; e < E; e += nwaves) {
    const int s = src[e];
    const int d = dst[e];
    const float4 v = ((const float4*)(P + (size_t)s * DIM))[lane];
    float* out = agg + (size_t)d * DIM + lane * 4;
    unsafeAtomicAdd(out + 0, v.x);
    unsafeAtomicAdd(out + 1, v.y);
    unsafeAtomicAdd(out + 2, v.z);
    unsafeAtomicAdd(out + 3, v.w);
  }
}

// ---------------------------------------------------------------------------
// h = relu(agg / max(deg,1) + bias)   (in-place capable: out may alias agg)
// ---------------------------------------------------------------------------
__global__ void __launch_bounds__(256)
finalize(const float* __restrict__ agg, const float* __restrict__ deg,
         const float* __restrict__ bias, float* __restrict__ out, int N) {
  const size_t i = (size_t)blockIdx.x * blockDim.x + threadIdx.x;
  if (i >= (size_t)N * DIM) return;
  const int node = (int)(i >> 7);
  const int c    = (int)(i & (DIM - 1));
  const float dg = fmaxf(deg[node], 1.0f);
  const float h  = agg[i] / dg + bias[c];
  out[i] = fmaxf(h, 0.0f);
}

// ---------------------------------------------------------------------------
extern "C" void kernel_launch(void* const* d_in, const int* in_sizes, int n_in,
                              void* d_out, int out_size, void* d_ws, size_t ws_size,
                              hipStream_t stream) {
  const float* x_user = (const float*)d_in[0];
  const float* x_item = (const float*)d_in[1];
  const float* W1_ui  = (const float*)d_in[2];
  const float* b1_ui  = (const float*)d_in[3];
  const float* W1_iu  = (const float*)d_in[4];
  const float* b1_iu  = (const float*)d_in[5];
  const float* W2_ui  = (const float*)d_in[6];
  const float* b2_ui  = (const float*)d_in[7];
  const float* W2_iu  = (const float*)d_in[8];
  const float* b2_iu  = (const float*)d_in[9];
  const int* e_ui_src = (const int*)d_in[10];
  const int* e_ui_dst = (const int*)d_in[11];
  const int* e_iu_src = (const int*)d_in[12];
  const int* e_iu_dst = (const int*)d_in[13];

  const int N = in_sizes[0] / DIM;   // NU == NI == 100000
  const int E = in_sizes[10];        // 1,000,000
  const size_t nd = (size_t)N * DIM;

  float* Pa     = (float*)d_ws;          // 51.2 MB projection buffer
  float* Pb     = Pa + nd;               // 51.2 MB projection buffer
  float* deg_ui = Pb + nd;               // item in-degrees (N floats)
  float* deg_iu = deg_ui + N;            // user in-degrees (N floats)

  float* out_user = (float*)d_out;       // final h_user2; temp h_user
  float* out_item = out_user + nd;       // final h_item2; temp h_item

  const int gemm_blocks = (N + 127) / 128;
  const int scat_blocks = 2048;
  const int fin_blocks  = (int)((nd + 255) / 256);

  // Degrees are layer-invariant per edge type: count once.
  hipMemsetAsync(deg_ui, 0, N * sizeof(float), stream);
  hipMemsetAsync(deg_iu, 0, N * sizeof(float), stream);
  count_deg<<<scat_blocks, 256, 0, stream>>>(e_ui_dst, deg_ui, E);
  count_deg<<<scat_blocks, 256, 0, stream>>>(e_iu_dst, deg_iu, E);

  // ---- Layer 1: user -> item -------------------------------------------
  gemm_xw_wmma<<<gemm_blocks, 256, 0, stream>>>(x_user, W1_ui, Pa, N);
  hipMemsetAsync(out_item, 0, nd * sizeof(float), stream);
  scatter_rows<<<scat_blocks, 256, 0, stream>>>(Pa, e_ui_src, e_ui_dst, out_item, E);
  finalize<<<fin_blocks, 256, 0, stream>>>(out_item, deg_ui, b1_ui, out_item, N); // h_item

  // ---- Layer 1: item -> user -------------------------------------------
  gemm_xw_wmma<<<gemm_blocks, 256, 0, stream>>>(x_item, W1_iu, Pa, N);
  hipMemsetAsync(out_user, 0, nd * sizeof(float), stream);
  scatter_rows<<<scat_blocks, 256, 0, stream>>>(Pa, e_iu_src, e_iu_dst, out_user, E);
  finalize<<<fin_blocks, 256, 0, stream>>>(out_user, deg_iu, b1_iu, out_user, N); // h_user

  // ---- Layer 2 projections (before clobbering d_out halves) ------------
  gemm_xw_wmma<<<gemm_blocks, 256, 0, stream>>>(out_user, W2_ui, Pa, N); // h_user @ W2_ui
  gemm_xw_wmma<<<gemm_blocks, 256, 0, stream>>>(out_item, W2_iu, Pb, N); // h_item @ W2_iu

  // ---- Layer 2: user -> item (h_item2 into d_out item half) ------------
  hipMemsetAsync(out_item, 0, nd * sizeof(float), stream);
  scatter_rows<<<scat_blocks, 256, 0, stream>>>(Pa, e_ui_src, e_ui_dst, out_item, E);
  finalize<<<fin_blocks, 256, 0, stream>>>(out_item, deg_ui, b2_ui, out_item, N);

  // ---- Layer 2: item -> user (h_user2 into d_out user half) ------------
  hipMemsetAsync(out_user, 0, nd * sizeof(float), stream);
  scatter_rows<<<scat_blocks, 256, 0, stream>>>(Pb, e_iu_src, e_iu_dst, out_user, E);
  finalize<<<fin_blocks, 256, 0, stream>>>(out_user, deg_iu, b2_iu, out_user, N);
}